// BlockLinear_420906795595
// MI455X (gfx1250) — compile-verified
//
#include <hip/hip_runtime.h>
#include <hip/hip_bf16.h>

typedef float v2f __attribute__((ext_vector_type(2)));
typedef float v8f __attribute__((ext_vector_type(8)));

#define NBLK  64
#define BLK   64
#define DCOL  (NBLK * BLK)   // 4096 columns
#define TROWS 16384

// Block-diagonal GEMM, fp32, bandwidth-bound (~537 MB @ 23.3 TB/s => ~23us floor).
// Wave-level tiling:
//   - each wave owns (block n, 16-wide col tile c), holds B (64x16 WMMA-layout, 32 VGPRs)
//   - iterates 8 row-stripes of 16 rows: 16x global_load_b64 for A, 16x
//     v_wmma_f32_16x16x4_f32 (K=64 in steps of 4), 8x b32 stores for the 16x16 C tile.
// Grid: 64 blocks * 4 col-tiles * 16 row-chunks = 4096 WGs of 256 threads (8 waves).
// Row coverage per (n,c): 16 chunks * 8 waves * 8 stripes * 16 rows = 16384. 

__global__ __launch_bounds__(256) void blocklinear_wmma_f32(
    const float* __restrict__ inp,   // [16384, 4096]
    const float* __restrict__ wts,   // [64, 64, 64]
    float* __restrict__ out)         // [16384, 4096]
{
  const int wg    = blockIdx.x;
  const int n     = wg >> 6;          // 0..63  block index
  const int c     = (wg >> 4) & 3;    // 0..3   16-col tile within block
  const int chunk = wg & 15;          // 0..15  row chunk
  const int wave  = threadIdx.x >> 5; // 0..7
  const int lane  = threadIdx.x & 31;
  const int lo    = lane & 15;        // column / row-in-stripe selector
  const int hi    = lane >> 4;        // 0 or 1 (K-pair selector)

  // ---- Load B tile for (n,c) once, in WMMA B-matrix layout ----
  // Bv[j] holds K = 4*j + hi*2 (+1 for .y), N = lo  (B is 4x16 per WMMA step)
  const float* wbase = wts + n * (BLK * BLK) + c * 16 + lo;
  v2f Bv[16];
#pragma unroll
  for (int j = 0; j < 16; ++j) {
    const int k = 4 * j + hi * 2;
    v2f b;
    b.x = wbase[(k + 0) * BLK];
    b.y = wbase[(k + 1) * BLK];
    Bv[j] = b;
  }

  const int colbase = n * BLK;
  const int stripe0 = chunk * 64 + wave * 8;

#pragma unroll 1
  for (int i = 0; i < 8; ++i) {
    const int row0 = (stripe0 + i) * 16;

    // A tile: lane -> row (row0+lo), K offset hi*2; .x/.y are contiguous K values
    const float* arow = inp + (size_t)(row0 + lo) * DCOL + colbase + hi * 2;

    if (i < 7) {
      // prefetch next stripe's A row segment (64 floats = 2 cachelines)
      __builtin_prefetch(arow + (size_t)16 * DCOL, 0, 3);
      __builtin_prefetch(arow + (size_t)16 * DCOL + 32, 0, 3);
    }

    v2f Av[16];
#pragma unroll
    for (int j = 0; j < 16; ++j) {
      Av[j] = *(const v2f*)(arow + 4 * j);   // global_load_b64, 8B aligned
    }

    v8f C = {};
#pragma unroll
    for (int j = 0; j < 16; ++j) {
      C = __builtin_amdgcn_wmma_f32_16x16x4_f32(
          /*neg_a=*/false, Av[j],
          /*neg_b=*/false, Bv[j],
          /*c_mod=*/(short)0, C,
          /*reuse_a=*/false, /*reuse_b=*/false);
    }

    // C/D layout: VGPR v -> row (row0 + v + hi*8), col (colbase + c*16 + lo)
    float* orow = out + (size_t)(row0 + hi * 8) * DCOL + colbase + c * 16 + lo;
#pragma unroll
    for (int v = 0; v < 8; ++v) {
      orow[(size_t)v * DCOL] = C[v];
    }
  }
}

extern "C" void kernel_launch(void* const* d_in, const int* in_sizes, int n_in,
                              void* d_out, int out_size, void* d_ws, size_t ws_size,
                              hipStream_t stream) {
  const float* inp = (const float*)d_in[0];   // [16384, 4096] fp32
  const float* wts = (const float*)d_in[1];   // [64, 64, 64]   fp32
  float* out = (float*)d_out;                 // [16384, 4096] fp32

  dim3 grid(64 * 4 * 16);   // 4096 workgroups
  dim3 block(256);          // 8 waves (wave32)
  hipLaunchKernelGGL(blocklinear_wmma_f32, grid, block, 0, stream, inp, wts, out);
}